// LSTMFromScratch_28114855919929
// MI455X (gfx1250) — compile-verified
//
#include <hip/hip_runtime.h>
#include <hip/hip_bf16.h>

// ---------------------------------------------------------------------------
// LSTM forward for MI455X (gfx1250, wave32, WMMA).
//
//   Phase 1 (parallel):  G[b*S+s, 4H] = x[b,s,:] @ W + bias      (fp32 WMMA)
//   Phase 2 (sequential):per t: gates = G_t + h_{t-1} @ U;       (fp32 WMMA)
//                        c = sig(f)*c + sig(i)*tanh(g); h = sig(o)*tanh(c)
//
// fp32 WMMA (V_WMMA_F32_16X16X4_F32) keeps exact reference precision while
// using the matrix pipes. U (16MB) stays hot in the 192MB L2 across all 512
// steps; h ping-pongs through L2 between the 32 persistent workgroups, which
// synchronize once per timestep with a device-wide atomic barrier.
//
// B/U tiles are staged into LDS *pair-interleaved* ([k/2][col] = {rowK,rowK+1})
// so every WMMA B fragment is a single contiguous ds_load_b64 into an
// even-aligned VGPR pair (no v_dual_mov shuffles in the serial hot loop).
// ---------------------------------------------------------------------------

typedef __attribute__((ext_vector_type(2))) float v2f;
typedef __attribute__((ext_vector_type(8))) float v8f;

#define B_   64
#define S_   512
#define I_   1024
#define H_   1024
#define G4_  4096   // 4*H
#define NWG_REC 32  // persistent workgroups in phase 2 (each owns 32 hidden units)

// pitches (in elements) with pad to de-conflict LDS banks
#define AP_  36     // As row pitch (floats)   -> 144B, 16B aligned
#define HP_  68     // hs row pitch (floats)      -> 272B, 16B aligned
#define BP_  130    // pair-tile row pitch (float2) -> 1040B, 16B aligned

// ---------------------------------------------------------------------------
// Phase 1: G = x @ W + bias.   M=32768, N=4096, K=1024.
// Workgroup tile 64(M) x 128(N), K-chunk 32, 256 threads (8 waves).
// Wave w: M-tile (w&3)*16, N-strip (w>>2)*64 -> 4 accumulators of 16x16.
// ---------------------------------------------------------------------------
__global__ __launch_bounds__(256)
void gemm_xw_kernel(const float* __restrict__ x, const float* __restrict__ W,
                    const float* __restrict__ bias, float* __restrict__ G) {
    const int mBase = blockIdx.y * 64;
    const int nBase = blockIdx.x * 128;
    const int tid  = threadIdx.x;
    const int wave = tid >> 5;
    const int lane = tid & 31;
    const int lh = lane & 15;        // lane within half-wave
    const int hi = lane >> 4;        // half-wave select (K pair 0/1 vs 2/3)
    const int m_off = (wave & 3) * 16;
    const int n_off = (wave >> 2) * 64;

    __shared__ __align__(16) float  As[64][AP_];     // 64 x 32 (+pad), row-major
    __shared__ __align__(16) float2 Bs2[16][BP_];    // pair-interleaved 32 x 128

    v8f acc[4];
#pragma unroll
    for (int t = 0; t < 4; ++t)
#pragma unroll
        for (int r = 0; r < 8; ++r) acc[t][r] = 0.0f;

    for (int k0 = 0; k0 < I_; k0 += 32) {
        __syncthreads();
        // Stage A: 64 rows x 8 float4  (512 float4 / 256 threads)
#pragma unroll
        for (int i = 0; i < 2; ++i) {
            int idx = tid + i * 256;
            int row = idx >> 3;
            int c4  = idx & 7;
            float4 v = *(const float4*)(x + (size_t)(mBase + row) * I_ + k0 + c4 * 4);
            *(float4*)(&As[row][c4 * 4]) = v;
        }
        // Stage B pair-interleaved: 16 k-pairs x 32 col-groups (512 tasks / 256 thr)
#pragma unroll
        for (int i = 0; i < 2; ++i) {
            int idx = tid + i * 256;
            int kp  = idx >> 5;          // 0..15
            int col = (idx & 31) * 4;    // 0..124
            const float* p0 = W + (size_t)(k0 + 2 * kp) * G4_ + nBase + col;
            float4 e = *(const float4*)(p0);         // even row (K = 2kp)
            float4 o = *(const float4*)(p0 + G4_);   // odd  row (K = 2kp+1)
            float4 w0 = {e.x, o.x, e.y, o.y};
            float4 w1 = {e.z, o.z, e.w, o.w};
            *(float4*)(&Bs2[kp][col])     = w0;
            *(float4*)(&Bs2[kp][col + 2]) = w1;
        }
        __syncthreads();

#pragma unroll
        for (int k = 0; k < 32; k += 4) {
            // A fragment (16x4): lanes0-15 K k,k+1; lanes16-31 K k+2,k+3
            v2f a;
            a.x = As[m_off + lh][k + 2 * hi];
            a.y = As[m_off + lh][k + 2 * hi + 1];
            const int kp = (k >> 1) + hi;
#pragma unroll
            for (int t = 0; t < 4; ++t) {
                v2f b = *(const v2f*)(&Bs2[kp][n_off + t * 16 + lh]);  // one b64
                acc[t] = __builtin_amdgcn_wmma_f32_16x16x4_f32(
                    false, a, false, b, (short)0, acc[t], false, false);
            }
        }
    }

    // Epilogue: += bias, store to G.
#pragma unroll
    for (int t = 0; t < 4; ++t) {
        int col = nBase + n_off + t * 16 + lh;
        float bv = bias[col];
#pragma unroll
        for (int r = 0; r < 8; ++r) {
            int row = mBase + m_off + r + 8 * hi;   // C reg r: M=r (lo half) / r+8 (hi half)
            G[(size_t)row * G4_ + col] = acc[t][r] + bv;
        }
    }
}

// ---------------------------------------------------------------------------
// Device-wide sense barrier (bar[0]=count, bar[1]=generation).
// ---------------------------------------------------------------------------
__device__ __forceinline__ void grid_barrier(int* bar, int nwg) {
    __syncthreads();
    __threadfence();
    if (threadIdx.x == 0) {
        volatile int* vgen = bar + 1;
        int g0 = *vgen;
        int prev = atomicAdd(bar, 1);
        if (prev == nwg - 1) {
            *bar = 0;
            __threadfence();
            atomicAdd(bar + 1, 1);
        } else {
            while (*vgen == g0) { __builtin_amdgcn_s_sleep(1); }
        }
    }
    __syncthreads();
    __threadfence();
}

__device__ __forceinline__ float sigmoidf_(float v) {
    return 1.0f / (1.0f + __expf(-v));
}

// ---------------------------------------------------------------------------
// Phase 2: persistent recurrent kernel. 32 WGs x 256 threads.
// WG n owns hidden columns j0=n*32 .. j0+31 and the 4 gate blocks for them
// (128 gate columns total). Per step: 64x128x1024 GEMM (h@U) on fp32 WMMA,
// K staged through LDS in 64-chunks; U block staged pair-interleaved (L2-hot).
// c lives in registers (8 per thread) for the whole sequence; h is
// double-buffered in global so step t+1 reads while t+1 writes disjoint slot.
// ---------------------------------------------------------------------------
__global__ __launch_bounds__(256)
void lstm_rec_kernel(const float* __restrict__ G, const float* __restrict__ U,
                     float* __restrict__ hb, int* __restrict__ bar,
                     float* __restrict__ out) {
    const int j0   = blockIdx.x * 32;
    const int tid  = threadIdx.x;
    const int wave = tid >> 5;
    const int lane = tid & 31;
    const int lh = lane & 15;
    const int hi = lane >> 4;
    const int g  = wave >> 1;        // gate block 0..3 (f,i,o,g)
    const int nh = wave & 1;         // 16-col half within the 32-wide slice
    const int colBase = g * H_ + j0 + nh * 16;   // global gate column base
    const int lc = g * 32 + nh * 16 + lh;        // local U/gate column

    // GEMM staging and gate exchange reuse the same LDS (disjoint phases).
    __shared__ __align__(16) union {
        struct {
            float  hs[64][HP_];      // h chunk 64 x 64, row-major
            float2 Us2[32][BP_];     // U chunk pair-interleaved 64 x 128
        } s;
        float gates[4][B_][33];
    } smem;

    float c_reg[8];
#pragma unroll
    for (int p = 0; p < 8; ++p) c_reg[p] = 0.0f;

    float* hfin = out + (size_t)B_ * S_ * H_;
    float* cfin = hfin + (size_t)B_ * H_;

    for (int t = 0; t < S_; ++t) {
        v8f acc[4];
#pragma unroll
        for (int m = 0; m < 4; ++m)
#pragma unroll
            for (int r = 0; r < 8; ++r) acc[m][r] = 0.0f;

        if (t > 0) {   // h_{-1} == 0 -> skip GEMM at t==0
            const float* hprev = hb + (size_t)((t - 1) & 1) * (B_ * H_);
            for (int kc = 0; kc < H_; kc += 64) {
                __syncthreads();
                // Stage h chunk: 64 x 64  (1024 float4 / 256 threads)
#pragma unroll
                for (int i = 0; i < 4; ++i) {
                    int idx = tid + i * 256;
                    int row = idx >> 4;
                    int c4  = idx & 15;
                    float4 v = *(const float4*)(hprev + (size_t)row * H_ + kc + c4 * 4);
                    *(float4*)(&smem.s.hs[row][c4 * 4]) = v;
                }
                // Stage U chunk pair-interleaved: 32 k-pairs x 32 col-groups
                // (1024 tasks / 256 threads)
#pragma unroll
                for (int i = 0; i < 4; ++i) {
                    int idx = tid + i * 256;
                    int kp  = idx >> 5;          // 0..31
                    int col = (idx & 31) * 4;    // local col 0..124
                    int gg  = col >> 5;          // gate block
                    int jj  = col & 31;
                    const float* p0 = U + (size_t)(kc + 2 * kp) * G4_
                                        + gg * H_ + j0 + jj;
                    float4 e = *(const float4*)(p0);         // K = kc+2kp
                    float4 o = *(const float4*)(p0 + G4_);   // K = kc+2kp+1
                    float4 w0 = {e.x, o.x, e.y, o.y};
                    float4 w1 = {e.z, o.z, e.w, o.w};
                    *(float4*)(&smem.s.Us2[kp][col])     = w0;
                    *(float4*)(&smem.s.Us2[kp][col + 2]) = w1;
                }
                __syncthreads();
#pragma unroll
                for (int k = 0; k < 64; k += 4) {
                    const int kp = (k >> 1) + hi;
                    v2f b = *(const v2f*)(&smem.s.Us2[kp][lc]);   // one b64
#pragma unroll
                    for (int m = 0; m < 4; ++m) {
                        v2f a;
                        a.x = smem.s.hs[m * 16 + lh][k + 2 * hi];
                        a.y = smem.s.hs[m * 16 + lh][k + 2 * hi + 1];
                        acc[m] = __builtin_amdgcn_wmma_f32_16x16x4_f32(
                            false, a, false, b, (short)0, acc[m], false, false);
                    }
                }
            }
        }

        __syncthreads();   // done reading hs/Us -> safe to overwrite with gates
        // gates = acc + G_t  -> LDS, organized [gate][batch][j] for recombination
#pragma unroll
        for (int m = 0; m < 4; ++m) {
#pragma unroll
            for (int r = 0; r < 8; ++r) {
                int brow = m * 16 + r + 8 * hi;   // batch row 0..63
                float val = acc[m][r] +
                    G[((size_t)brow * S_ + t) * G4_ + colBase + lh];
                smem.gates[g][brow][nh * 16 + lh] = val;
            }
        }
        __syncthreads();

        // Elementwise LSTM cell for this WG's 64x32 (batch x hidden) block.
        float* hcur = hb + (size_t)(t & 1) * (B_ * H_);
#pragma unroll
        for (int p = 0; p < 8; ++p) {
            int idx = tid + p * 256;      // 0..2047
            int b = idx >> 5;
            int j = idx & 31;
            float f  = sigmoidf_(smem.gates[0][b][j]);
            float i_ = sigmoidf_(smem.gates[1][b][j]);
            float o  = sigmoidf_(smem.gates[2][b][j]);
            float gg = tanhf(smem.gates[3][b][j]);
            float c  = f * c_reg[p] + i_ * gg;
            float h  = o * tanhf(c);
            c_reg[p] = c;
            hcur[(size_t)b * H_ + j0 + j] = h;
            out[((size_t)b * S_ + t) * H_ + j0 + j] = h;
            if (t == S_ - 1) {
                hfin[(size_t)b * H_ + j0 + j] = h;
                cfin[(size_t)b * H_ + j0 + j] = c;
            }
        }

        grid_barrier(bar, (int)gridDim.x);   // publish h_t to all WGs
    }
}

__global__ void init_bar_kernel(int* bar) { bar[0] = 0; bar[1] = 0; }

// ---------------------------------------------------------------------------
// Launch
// ---------------------------------------------------------------------------
extern "C" void kernel_launch(void* const* d_in, const int* in_sizes, int n_in,
                              void* d_out, int out_size, void* d_ws, size_t ws_size,
                              hipStream_t stream) {
    const float* x    = (const float*)d_in[0];   // (64,512,1024)
    const float* W    = (const float*)d_in[1];   // (1024,4096)
    const float* U    = (const float*)d_in[2];   // (1024,4096)
    const float* bias = (const float*)d_in[3];   // (4096,)
    float* out = (float*)d_out;                  // (B,S,H) ++ (B,H) ++ (B,H)

    // Workspace: G (B*S x 4H) | h double-buffer (2 x B x H) | barrier (2 ints)
    float* G  = (float*)d_ws;
    float* hb = G + (size_t)B_ * S_ * G4_;
    int*  bar = (int*)(hb + (size_t)2 * B_ * H_);

    init_bar_kernel<<<1, 1, 0, stream>>>(bar);

    dim3 gridA(G4_ / 128, (B_ * S_) / 64);       // 32 x 512 WGs
    gemm_xw_kernel<<<gridA, 256, 0, stream>>>(x, W, bias, G);

    lstm_rec_kernel<<<NWG_REC, 256, 0, stream>>>(G, U, hb, bar, out);
}